// BootstrappedLoss_39350490366132
// MI455X (gfx1250) — compile-verified
//
#include <hip/hip_runtime.h>
#include <stdint.h>

// Problem constants (from reference)
#define B_    8
#define C_    19
#define H_    512
#define W_    1024
#define HWP   (H_ * W_)        // 524288 pixels per batch
#define KSEL  (HWP / 4)        // 131072 = top 25%
#define NF4   (HWP / 4)        // float4 slots per batch-plane

// Streaming pass geometry
#define PASSBLK 64                  // blocks per batch for streaming passes
#define SEG     (HWP / PASSBLK)     // 8192 u32 per block
#define CHUNK   1024                // u32 per pipeline stage (256 lanes * 4)
#define NITER   (SEG / CHUNK)       // 8 double-buffered iterations

// native clang vectors (required by __builtin_nontemporal_load)
typedef __attribute__((ext_vector_type(4))) float    f32x4;
typedef __attribute__((ext_vector_type(4))) int      i32x4;
typedef __attribute__((ext_vector_type(4))) unsigned u32x4;

// ---- workspace layout (bytes) ----
static constexpr size_t OFF_LOSS = 0;                                   // B*HWP u32  (16.78 MB)
static constexpr size_t OFF_H0   = OFF_LOSS + (size_t)B_ * HWP * 4;     // B*2048 u32
static constexpr size_t OFF_H1   = OFF_H0   + (size_t)B_ * 2048 * 4;    // B*2048 u32
static constexpr size_t OFF_H2   = OFF_H1   + (size_t)B_ * 2048 * 4;    // B*1024 u32
static constexpr size_t OFF_ST   = OFF_H2   + (size_t)B_ * 1024 * 4;    // B*8 u32: [0]=prefix [1]=krem [2]=tbits [3]=r
static constexpr size_t OFF_PART = OFF_ST   + (size_t)B_ * 8 * 4;       // PASSBLK*B_ f32
// total ~16.94 MB

// ---------- CDNA5 async global->LDS path (ASYNCcnt) ----------
__device__ __forceinline__ void async_ld_b128(unsigned lds_off, const void* g) {
  asm volatile("global_load_async_to_lds_b128 %0, %1, off"
               :: "v"(lds_off), "v"((unsigned long long)g)
               : "memory");
}
__device__ __forceinline__ void wait_async_le1() {
  asm volatile("s_wait_asynccnt 1" ::: "memory");
}
__device__ __forceinline__ void wait_async_le0() {
  asm volatile("s_wait_asynccnt 0" ::: "memory");
}
__device__ __forceinline__ unsigned lds_offset_of(const void* p) {
  // generic pointer to __shared__: low 32 bits are the workgroup-relative LDS offset
  return (unsigned)(size_t)p;
}

// ---------- kernel: deterministic partial sums of values > threshold ----------
// (defined first so the disasm snippet shows the async global->LDS pipeline)
__global__ __launch_bounds__(256) void k_final_partial(
    const unsigned* __restrict__ lossBits, const unsigned* __restrict__ state,
    float* __restrict__ partials) {
  __shared__ __align__(16) unsigned stage[2][CHUNK];
  __shared__ float red[256];
  const int b = blockIdx.y;
  const int t = threadIdx.x;
  const unsigned tb = state[b * 8 + 2];
  const unsigned* src = lossBits + (size_t)b * HWP + (size_t)blockIdx.x * SEG;
  const unsigned lds0 = lds_offset_of(&stage[0][t * 4]);
  const unsigned lds1 = lds_offset_of(&stage[1][t * 4]);

  float acc = 0.f;
  async_ld_b128(lds0, src + t * 4);
  for (int it = 0; it < NITER; ++it) {
    if (it + 1 < NITER) {
      async_ld_b128((it & 1) ? lds0 : lds1, src + (it + 1) * CHUNK + t * 4);
      wait_async_le1();   // in-order completion: oldest (current) chunk is done
    } else {
      wait_async_le0();
    }
    const unsigned* buf = (it & 1) ? &stage[1][t * 4] : &stage[0][t * 4];
    u32x4 x = *(const u32x4*)buf;
    unsigned xs[4] = {x.x, x.y, x.z, x.w};
#pragma unroll
    for (int j = 0; j < 4; ++j)
      if (xs[j] > tb) acc += __uint_as_float(xs[j]);
  }
  red[t] = acc;
  __syncthreads();
  for (int off = 128; off > 0; off >>= 1) {
    if (t < off) red[t] += red[t + off];
    __syncthreads();
  }
  if (t == 0) partials[b * gridDim.x + blockIdx.x] = red[0];
}

// ---------- kernel: refinement histogram over L2-resident loss bits ----------
__global__ __launch_bounds__(256) void k_hist_pass(
    const unsigned* __restrict__ lossBits, const unsigned* __restrict__ state,
    unsigned* __restrict__ hist, int level) {
  __shared__ unsigned lh[2048];
  __shared__ __align__(16) unsigned stage[2][CHUNK];
  const int b = blockIdx.y;
  const int t = threadIdx.x;
  const int nbins = (level == 1) ? 2048 : 1024;
  for (int i = t; i < nbins; i += 256) lh[i] = 0u;
  __syncthreads();

  const unsigned prefix = state[b * 8 + 0];
  const unsigned* src = lossBits + (size_t)b * HWP + (size_t)blockIdx.x * SEG;
  const unsigned lds0 = lds_offset_of(&stage[0][t * 4]);
  const unsigned lds1 = lds_offset_of(&stage[1][t * 4]);

  async_ld_b128(lds0, src + t * 4);
  for (int it = 0; it < NITER; ++it) {
    if (it + 1 < NITER) {
      async_ld_b128((it & 1) ? lds0 : lds1, src + (it + 1) * CHUNK + t * 4);
      wait_async_le1();
    } else {
      wait_async_le0();
    }
    const unsigned* buf = (it & 1) ? &stage[1][t * 4] : &stage[0][t * 4];
    u32x4 x = *(const u32x4*)buf;
    unsigned xs[4] = {x.x, x.y, x.z, x.w};
#pragma unroll
    for (int j = 0; j < 4; ++j) {
      unsigned v = xs[j];
      if (level == 1) {
        if ((v >> 21) == prefix) atomicAdd(&lh[(v >> 10) & 0x7FFu], 1u);
      } else {
        if ((v >> 10) == prefix) atomicAdd(&lh[v & 0x3FFu], 1u);
      }
    }
  }
  __syncthreads();
  for (int i = t; i < nbins; i += 256) {
    unsigned c = lh[i];
    if (c) atomicAdd(&hist[b * nbins + i], c);
  }
}

// ---------- kernel: zero histograms + state ----------
__global__ void k_zero(unsigned* __restrict__ p, int n) {
  int i = blockIdx.x * blockDim.x + threadIdx.x;
  if (i < n) p[i] = 0u;
}

// ---------- kernel: fused CE loss + store bits + level-0 histogram ----------
__global__ __launch_bounds__(256) void k_loss_hist0(
    const float* __restrict__ logits, const int* __restrict__ labels,
    unsigned* __restrict__ lossBits, unsigned* __restrict__ hist0) {
  __shared__ unsigned lh[2048];
  const int b = blockIdx.y;
  for (int i = threadIdx.x; i < 2048; i += 256) lh[i] = 0u;
  __syncthreads();

  const int q = blockIdx.x * 256 + threadIdx.x;  // float4 index within batch plane
  const f32x4* lg4 = (const f32x4*)(logits + (size_t)b * C_ * HWP);
  const i32x4* lb4 = (const i32x4*)(labels + (size_t)b * HWP);
  u32x4*       ob4 = (u32x4*)(lossBits + (size_t)b * HWP);

  int lab[4];
  {
    i32x4 L = __builtin_nontemporal_load(&lb4[q]);  // single-use stream: keep out of caches
    lab[0] = L.x; lab[1] = L.y; lab[2] = L.z; lab[3] = L.w;
  }

  float v[C_][4];
  float m[4] = {-3.4e38f, -3.4e38f, -3.4e38f, -3.4e38f};
#pragma unroll
  for (int c = 0; c < C_; ++c) {
    f32x4 x = __builtin_nontemporal_load(&lg4[(size_t)c * NF4 + q]);  // 318 MB read-once stream
    v[c][0] = x.x; v[c][1] = x.y; v[c][2] = x.z; v[c][3] = x.w;
#pragma unroll
    for (int j = 0; j < 4; ++j) m[j] = fmaxf(m[j], v[c][j]);
  }
  float s[4] = {0.f, 0.f, 0.f, 0.f};
  float g[4] = {0.f, 0.f, 0.f, 0.f};
#pragma unroll
  for (int c = 0; c < C_; ++c) {
#pragma unroll
    for (int j = 0; j < 4; ++j) {
      s[j] += __expf(v[c][j] - m[j]);
      g[j] = (lab[j] == c) ? v[c][j] : g[j];
    }
  }
  unsigned bits[4];
#pragma unroll
  for (int j = 0; j < 4; ++j) {
    float loss = m[j] + __logf(s[j]) - g[j];   // CE >= 0, so uint order == float order
    bits[j] = __float_as_uint(loss);
  }
  u32x4 ob = {bits[0], bits[1], bits[2], bits[3]};
  ob4[q] = ob;   // regular-temporal store: 16.8 MB stays L2-resident for refinement passes
#pragma unroll
  for (int j = 0; j < 4; ++j) atomicAdd(&lh[bits[j] >> 21], 1u);

  __syncthreads();
  for (int i = threadIdx.x; i < 2048; i += 256) {
    unsigned c = lh[i];
    if (c) atomicAdd(&hist0[b * 2048 + i], c);
  }
}

// ---------- kernel: radix digit selection (one block per batch) ----------
__global__ __launch_bounds__(256) void k_select(
    const unsigned* __restrict__ hist, unsigned* __restrict__ state,
    int nbins, int level) {
  const int b = blockIdx.x;
  const int t = threadIdx.x;
  __shared__ unsigned sh[2048];
  __shared__ unsigned ss[256];
  __shared__ unsigned selD, selG;
  const unsigned* h = hist + b * nbins;
  const int chunk = nbins >> 8;   // 8 or 4 bins per thread

  unsigned local = 0;
  for (int i = 0; i < chunk; ++i) { unsigned x = h[t * chunk + i]; sh[t * chunk + i] = x; local += x; }
  ss[t] = local;
  __syncthreads();
  // inclusive suffix scan over per-thread chunk sums
  for (int off = 1; off < 256; off <<= 1) {
    unsigned add = (t + off < 256) ? ss[t + off] : 0u;
    __syncthreads();
    ss[t] += add;
    __syncthreads();
  }
  const unsigned k = (level == 0) ? (unsigned)KSEL : state[b * 8 + 1];
  unsigned cum = (t + 1 < 256) ? ss[t + 1] : 0u;  // count in bins strictly above my chunk
  for (int i = chunk - 1; i >= 0; --i) {
    unsigned idx   = (unsigned)(t * chunk + i);
    unsigned above = cum;       // elements strictly greater than bin idx
    cum += sh[idx];
    if (cum >= k && above < k) { selD = idx; selG = above; }  // unique bin block-wide
  }
  __syncthreads();
  if (t == 0) {
    unsigned digit = selD, G = selG;
    unsigned newk = k - G;      // still needed from within selected bin
    if (level == 0)      { state[b*8+0] = digit;                        state[b*8+1] = newk; }
    else if (level == 1) { state[b*8+0] = (state[b*8+0] << 11) | digit; state[b*8+1] = newk; }
    else                 { state[b*8+2] = (state[b*8+0] << 10) | digit; state[b*8+3] = newk; }
  }
}

// ---------- kernel: final reduce + tie handling + mean ----------
__global__ __launch_bounds__(256) void k_final_reduce(
    const float* __restrict__ partials, const unsigned* __restrict__ state,
    float* __restrict__ out, int nPart) {
  __shared__ float red[256];
  const int t = threadIdx.x;
  float s = 0.f;
  for (int i = t; i < nPart; i += 256) s += partials[i];
  red[t] = s;
  __syncthreads();
  for (int off = 128; off > 0; off >>= 1) {
    if (t < off) red[t] += red[t + off];
    __syncthreads();
  }
  if (t == 0) {
    float tot = red[0];
    for (int b = 0; b < B_; ++b)
      tot += (float)state[b * 8 + 3] * __uint_as_float(state[b * 8 + 2]);  // r * t (exact tie handling)
    out[0] = tot / (float)(B_ * KSEL);
  }
}

extern "C" void kernel_launch(void* const* d_in, const int* in_sizes, int n_in,
                              void* d_out, int out_size, void* d_ws, size_t ws_size,
                              hipStream_t stream) {
  (void)in_sizes; (void)n_in; (void)out_size; (void)ws_size;
  const float* logits = (const float*)d_in[0];
  const int*   labels = (const int*)d_in[1];
  float* out = (float*)d_out;
  char* ws = (char*)d_ws;
  unsigned* lossBits = (unsigned*)(ws + OFF_LOSS);
  unsigned* h0 = (unsigned*)(ws + OFF_H0);
  unsigned* h1 = (unsigned*)(ws + OFF_H1);
  unsigned* h2 = (unsigned*)(ws + OFF_H2);
  unsigned* st = (unsigned*)(ws + OFF_ST);
  float* parts = (float*)(ws + OFF_PART);

  // h0..state are contiguous: zero them all in one shot (fresh state every call)
  const int nz = B_ * (2048 + 2048 + 1024 + 8);
  k_zero<<<(nz + 255) / 256, 256, 0, stream>>>(h0, nz);

  k_loss_hist0<<<dim3(NF4 / 256, B_), 256, 0, stream>>>(logits, labels, lossBits, h0);
  k_select<<<B_, 256, 0, stream>>>(h0, st, 2048, 0);
  k_hist_pass<<<dim3(PASSBLK, B_), 256, 0, stream>>>(lossBits, st, h1, 1);
  k_select<<<B_, 256, 0, stream>>>(h1, st, 2048, 1);
  k_hist_pass<<<dim3(PASSBLK, B_), 256, 0, stream>>>(lossBits, st, h2, 2);
  k_select<<<B_, 256, 0, stream>>>(h2, st, 1024, 2);
  k_final_partial<<<dim3(PASSBLK, B_), 256, 0, stream>>>(lossBits, st, parts);
  k_final_reduce<<<1, 256, 0, stream>>>(parts, st, out, PASSBLK * B_);
}